// TwoSpeciesSystem_Diab_44341242364445
// MI455X (gfx1250) — compile-verified
//
#include <hip/hip_runtime.h>
#include <stdint.h>

#define TPB 256
#define NSWEEP 5

// ---- CDNA5 async global->LDS staging (ASYNCcnt-tracked, CDNA5 ISA) ----
__device__ __forceinline__ void async_ld_b128(uint32_t lds_byte_addr, const void* gptr) {
    // vdst = LDS byte address VGPR, vaddr = 64-bit global address VGPR pair, GV mode
    asm volatile("global_load_async_to_lds_b128 %0, %1, off"
                 :: "v"(lds_byte_addr), "v"((unsigned long long)(uintptr_t)gptr)
                 : "memory");
}

__device__ __forceinline__ void wait_asynccnt0() {
    asm volatile("s_wait_asynccnt 0" ::: "memory");
}

// ---- one Jacobi rotation in plane (P,Q) on a symmetric 4x4 kept in registers ----
// Uses raw TRANS-unit ops (v_sqrt/v_rcp/v_rsq, ~1 ulp) which co-execute with VALU.
template <int P, int Q>
__device__ __forceinline__ void jrot(float A[4][4]) {
    float apq = A[P][Q];
    float h   = 0.5f * (A[Q][Q] - A[P][P]);

    // |apq| threshold: below this the rotation is numerically identity for O(1)
    // spectra, and it guarantees den is a normal float before raw v_rcp_f32.
    bool  active = fabsf(apq) > 1e-20f;

    float r   = __builtin_amdgcn_sqrtf(__builtin_fmaf(h, h, apq * apq)); // v_sqrt_f32
    float den = fabsf(h) + r;                      // den >= |apq| > 1e-20 when active
    float rcp = __builtin_amdgcn_rcpf(den);        // v_rcp_f32, rcp > 0
    // t = sign(h) * apq / den, |t| <= 1  (rcp>0 so copysign folds the sign flip)
    float t   = active ? (apq * copysignf(rcp, h)) : 0.0f;

    float c   = __builtin_amdgcn_rsqf(__builtin_fmaf(t, t, 1.0f));       // v_rsq_f32
    float s   = t * c;

    float app = A[P][P], aqq = A[Q][Q];
    A[P][P] = __builtin_fmaf(-t, apq, app);
    A[Q][Q] = __builtin_fmaf( t, apq, aqq);
    A[P][Q] = 0.0f;
    A[Q][P] = 0.0f;
#pragma unroll
    for (int k = 0; k < 4; ++k) {
        if (k != P && k != Q) {
            float akp = A[k][P], akq = A[k][Q];
            float nkp = __builtin_fmaf(c, akp, -s * akq);
            float nkq = __builtin_fmaf(s, akp,  c * akq);
            A[k][P] = nkp; A[P][k] = nkp;
            A[k][Q] = nkq; A[Q][k] = nkq;
        }
    }
}

__device__ __forceinline__ void cswap(float& a, float& b) {
    float lo = fminf(a, b);
    float hi = fmaxf(a, b);
    a = lo; b = hi;
}

__global__ __launch_bounds__(TPB) void eig4_jacobi_kernel(
    const float* __restrict__ E1, const float* __restrict__ E2,
    const float* __restrict__ C,  float* __restrict__ out, int n)
{
    __shared__ float sE1[TPB * 2];   // 2 KB
    __shared__ float sE2[TPB * 2];   // 2 KB
    __shared__ float sC [TPB * 6];   // 6 KB  (10 KB total, trivial vs 320 KB/WGP)

    const int       t    = threadIdx.x;
    const long long base = (long long)blockIdx.x * TPB;
    const long long i    = base + t;
    const bool fullTile  = (base + TPB) <= (long long)n;

    float d0 = 0.f, d1 = 0.f, d2 = 0.f, d3 = 0.f;
    float c01 = 0.f, c02 = 0.f, c03 = 0.f, c12 = 0.f, c13 = 0.f, c23 = 0.f;

    if (fullTile) {
        const char* gE1 = (const char*)(E1 + base * 2);   // 2048 B = 128 x b128
        const char* gE2 = (const char*)(E2 + base * 2);   // 2048 B = 128 x b128
        const char* gC  = (const char*)(C  + base * 6);   // 6144 B = 384 x b128

        const uint32_t lE1 = (uint32_t)(uintptr_t)&sE1[0];
        const uint32_t lE2 = (uint32_t)(uintptr_t)&sE2[0];
        const uint32_t lC  = (uint32_t)(uintptr_t)&sC[0];

        // lanes 0..127 stage E1, lanes 128..255 stage E2
        if (t < 128) {
            async_ld_b128(lE1 + 16u * (uint32_t)t, gE1 + 16 * t);
        } else {
            const int u = t - 128;
            async_ld_b128(lE2 + 16u * (uint32_t)u, gE2 + 16 * u);
        }
        // C: 384 chunks over 256 lanes
        async_ld_b128(lC + 16u * (uint32_t)t, gC + 16 * t);
        if (t < 128) {
            const int u = t + 256;
            async_ld_b128(lC + 16u * (uint32_t)u, gC + 16 * u);
        }

        wait_asynccnt0();     // retire this wave's ASYNCcnt
        __syncthreads();      // make all waves' LDS writes visible

        d0  = sE1[2 * t + 0]; d1 = sE1[2 * t + 1];
        d2  = sE2[2 * t + 0]; d3 = sE2[2 * t + 1];
        // stride-6-dword reads: gcd(3,32)=1 -> bank-conflict-free across wave32
        c01 = sC[6 * t + 0];  c02 = sC[6 * t + 1];  c03 = sC[6 * t + 2];
        c12 = sC[6 * t + 3];  c13 = sC[6 * t + 4];  c23 = sC[6 * t + 5];
    } else {
        if (i < n) {
            d0 = E1[2 * i + 0]; d1 = E1[2 * i + 1];
            d2 = E2[2 * i + 0]; d3 = E2[2 * i + 1];
            const float* cp = C + 6 * i;
            c01 = cp[0]; c02 = cp[1]; c03 = cp[2];
            c12 = cp[3]; c13 = cp[4]; c23 = cp[5];
        }
    }

    if (i >= n) return;

    // Assemble symmetric H (triu order from np.triu_indices(4, k=1)):
    // (0,1)(0,2)(0,3)(1,2)(1,3)(2,3)
    float A[4][4];
    A[0][0] = d0; A[1][1] = d1; A[2][2] = d2; A[3][3] = d3;
    A[0][1] = c01; A[1][0] = c01;
    A[0][2] = c02; A[2][0] = c02;
    A[0][3] = c03; A[3][0] = c03;
    A[1][2] = c12; A[2][1] = c12;
    A[1][3] = c13; A[3][1] = c13;
    A[2][3] = c23; A[3][2] = c23;

    // Cyclic Jacobi: quadratic convergence; 5 sweeps >> fp32 precision for 4x4.
#pragma unroll 1
    for (int sw = 0; sw < NSWEEP; ++sw) {
        jrot<0, 1>(A); jrot<0, 2>(A); jrot<0, 3>(A);
        jrot<1, 2>(A); jrot<1, 3>(A); jrot<2, 3>(A);
    }

    float e0 = A[0][0], e1 = A[1][1], e2 = A[2][2], e3 = A[3][3];
    // 5-comparator sorting network, ascending
    cswap(e0, e1); cswap(e2, e3);
    cswap(e0, e2); cswap(e1, e3);
    cswap(e1, e2);

    float4 r; r.x = e0; r.y = e1; r.z = e2; r.w = e3;
    reinterpret_cast<float4*>(out)[i] = r;   // coalesced 16 B/lane store
}

extern "C" void kernel_launch(void* const* d_in, const int* in_sizes, int n_in,
                              void* d_out, int out_size, void* d_ws, size_t ws_size,
                              hipStream_t stream) {
    const float* E1 = (const float*)d_in[0];
    const float* E2 = (const float*)d_in[1];
    const float* C  = (const float*)d_in[2];
    float* out      = (float*)d_out;

    const int n      = in_sizes[2] / 6;           // batch size from C's flat count
    const int blocks = (n + TPB - 1) / TPB;

    hipLaunchKernelGGL(eig4_jacobi_kernel, dim3(blocks), dim3(TPB), 0, stream,
                       E1, E2, C, out, n);
}